// PitchEncoder_695784702073
// MI455X (gfx1250) — compile-verified
//
#include <hip/hip_runtime.h>
#include <hip/hip_bf16.h>

typedef __attribute__((ext_vector_type(2))) float v2f;
typedef __attribute__((ext_vector_type(8))) float v8f;

#define SRATE    48000
#define WINLEN   4800
#define HOPLEN   480
#define TAUMAX   2400
#define NFR_IN   391          // (192000-4800)/480 + 1
#define NFR_OUT  400
#define HID      256
#define NCHUNK   75           // 4800 / 64
#define NDT      39           // diag tiles: D = 16*dt + (n-m) covers [0,599]
#define XPAD     7296         // window + zero pad for max B read (7234)
#define THRESH   0.1f
#define BIGI     (1 << 30)

__device__ __forceinline__ v8f wmma_f32(v2f a, v2f b, v8f c) {
  // D = A(16x4) * B(4x16) + C(16x16), fp32
  return __builtin_amdgcn_wmma_f32_16x16x4_f32(
      false, a, false, b, (short)0, c, false, false);
}

// ---------------------------------------------------------------------------
// Kernel 1: one workgroup (8 waves) per (batch, frame).
// WMMA-tiled lagged autocorrelation + cumsum + CMDF + YIN search -> pitch.
// ---------------------------------------------------------------------------
__global__ __launch_bounds__(256) void yin_kernel(const float* __restrict__ y,
                                                  float* __restrict__ pitch) {
  __shared__ float xs[XPAD];       // window, later in-place cumsum of squares
  __shared__ float corr[TAUMAX];   // autocorr accumulators
  __shared__ float dbuf[TAUMAX];   // d1, then cmdf_full (2399 used)
  __shared__ float scan[256];
  __shared__ int s_fb, s_tau;

  const int blk  = blockIdx.x;
  const int b    = blk / NFR_IN;
  const int f    = blk % NFR_IN;
  const int tid  = threadIdx.x;
  const int lane = tid & 31;
  const int wv   = tid >> 5;

  const float* ybf = y + (size_t)b * 192000 + (size_t)f * HOPLEN;
  for (int i = tid; i < WINLEN; i += 256) xs[i] = ybf[i];
  for (int i = WINLEN + tid; i < XPAD; i += 256) xs[i] = 0.f;
  for (int i = tid; i < TAUMAX; i += 256) corr[i] = 0.f;
  if (tid == 0) { s_fb = BIGI; s_tau = BIGI; }
  __syncthreads();

  // ---- Phase 1: autocorrelation via f32 WMMA diagonal-band tiles ----
  // A[m][k] = xs[64c + 4m + k]            (lane: m=lane&15, k-half=lane>>4)
  // B_r[k][n] = xs[64c + 64dt + r + 4n + k],  r = 0..3 share one 5-float read.
  // C_r[m][n] accumulates lag = 4*(16*dt + n - m) + r over all 75 chunks.
  const int m = lane & 15;
  const int h = lane >> 4;          // 0: K={0,1}, 1: K={2,3}
  const int aoff = 4 * m + 2 * h;
  for (int dt = wv; dt < NDT; dt += 8) {   // wave-uniform -> EXEC all ones
    v8f acc0 = {}, acc1 = {}, acc2 = {}, acc3 = {};
    const int boff = 64 * dt + aoff;
    for (int c = 0; c < NCHUNK; ++c) {
      const int base = 64 * c;
      v2f a   = *(const v2f*)&xs[base + aoff];
      v2f b01 = *(const v2f*)&xs[base + boff];       // floats r+0, r+1 base
      v2f b23 = *(const v2f*)&xs[base + boff + 2];
      float b4 = xs[base + boff + 4];
      v2f br1; br1.x = b01.y; br1.y = b23.x;
      v2f br3; br3.x = b23.y; br3.y = b4;
      acc0 = wmma_f32(a, b01, acc0);
      acc1 = wmma_f32(a, br1, acc1);
      acc2 = wmma_f32(a, b23, acc2);
      acc3 = wmma_f32(a, br3, acc3);
    }
    // diagonal reduction: lane holds C[M][N], M = 8*h + v, N = m
#pragma unroll
    for (int v = 0; v < 8; ++v) {
      const int M = 8 * h + v;
      const int D = 16 * dt + m - M;     // lag = 4*D + r
      if (D >= 0 && D < 600) {
        if (D >= 1) atomicAdd(&corr[4 * D], acc0[v]);   // lag 0 not needed
        atomicAdd(&corr[4 * D + 1], acc1[v]);
        atomicAdd(&corr[4 * D + 2], acc2[v]);
        atomicAdd(&corr[4 * D + 3], acc3[v]);
      }
    }
  }
  __syncthreads();

  // ---- Phase 2: in-place cumulative sum of squares (sqrcs) ----
  {
    const int t0 = tid * 19;                       // 256*19 = 4864 >= 4800
    float s = 0.f;
    for (int i = 0; i < 19; ++i) {
      int t = t0 + i;
      if (t < WINLEN) { float xv = xs[t]; s += xv * xv; }
    }
    scan[tid] = s;
    __syncthreads();
    for (int off = 1; off < 256; off <<= 1) {
      float vs = scan[tid];
      if (tid >= off) vs += scan[tid - off];
      __syncthreads();
      scan[tid] = vs;
      __syncthreads();
    }
    float run = (tid == 0) ? 0.f : scan[tid - 1];
    for (int i = 0; i < 19; ++i) {
      int t = t0 + i;
      if (t < WINLEN) { float xv = xs[t]; run += xv * xv; xs[t] = run; }
    }
  }
  __syncthreads();

  // ---- Phase 3: d1 = diff[1:], then CMDF via second scan ----
  // sqrcs[i] = (i==0) ? 0 : xs[i-1];  E = sqrcs[WIN]
  const float E = xs[WINLEN - 1];
  {
    const int j0 = tid * 10;                       // 256*10 = 2560 >= 2399
    for (int i = 0; i < 10; ++i) {
      int j = j0 + i;
      if (j < TAUMAX - 1) {
        int tau = j + 1;                           // tau in [1, 2399]
        float d = E + xs[WINLEN - tau - 1] - xs[tau - 1] - 2.f * corr[tau];
        dbuf[j] = d;
      }
    }
  }
  __syncthreads();
  {
    const int j0 = tid * 10;
    float s = 0.f;
    for (int i = 0; i < 10; ++i) {
      int j = j0 + i;
      if (j < TAUMAX - 1) s += dbuf[j];
    }
    scan[tid] = s;
    __syncthreads();
    for (int off = 1; off < 256; off <<= 1) {
      float vs = scan[tid];
      if (tid >= off) vs += scan[tid - off];
      __syncthreads();
      scan[tid] = vs;
      __syncthreads();
    }
    float run = (tid == 0) ? 0.f : scan[tid - 1];
    for (int i = 0; i < 10; ++i) {
      int j = j0 + i;
      if (j < TAUMAX - 1) {
        float d = dbuf[j];
        run += d;
        dbuf[j] = d * (float)(j + 1) / fmaxf(run, 1e-5f);   // cmdf_full[j]
      }
    }
  }
  __syncthreads();

  // ---- Phase 4: YIN search on cmdf_s[t] = cmdf_full[t+2], t in [0,2397) ----
  const int NS = TAUMAX - 1 - 2;  // 2397
  for (int t = tid; t < NS; t += 256)
    if (dbuf[t + 2] < THRESH) atomicMin(&s_fb, t);
  __syncthreads();
  int fb = s_fb;
  if (fb <= 0 || fb >= BIGI) fb = TAUMAX;   // argmax==0 -> TAU_MAX (ref quirk)
  for (int t = tid; t < NS; t += 256) {
    if (t >= fb) {
      bool inc = (t == NS - 1) ? true : (dbuf[t + 3] - dbuf[t + 2] >= 0.f);
      if (inc) atomicMin(&s_tau, t);
    }
  }
  __syncthreads();
  if (tid == 0) {
    int tau = (s_tau >= BIGI) ? 0 : s_tau;
    float p = (tau > 0) ? (float)SRATE / (float)(tau + 2 + 1) : 0.f;
    pitch[b * NFR_IN + f] = p;
  }
}

// ---------------------------------------------------------------------------
// Kernel 2: linear resize 391->400, layer 1 (fan_in=1), LayerNorm, leaky ReLU.
// One 256-thread block per output row.
// ---------------------------------------------------------------------------
__global__ __launch_bounds__(256) void mlp1_kernel(
    const float* __restrict__ pitch, const float* __restrict__ w1,
    const float* __restrict__ b1, const float* __restrict__ g1,
    const float* __restrict__ bt1, float* __restrict__ out) {
  __shared__ float red[16];
  const int row = blockIdx.x;          // b*400 + fo
  const int b   = row / NFR_OUT;
  const int fo  = row % NFR_OUT;
  const int c   = threadIdx.x;

  float pos = (float)fo * ((float)(NFR_IN - 1) / (float)(NFR_OUT - 1));
  int i0 = (int)floorf(pos);
  if (i0 < 0) i0 = 0;
  if (i0 > NFR_IN - 2) i0 = NFR_IN - 2;
  float w = pos - (float)i0;
  float p = pitch[b * NFR_IN + i0] * (1.f - w) + pitch[b * NFR_IN + i0 + 1] * w;

  float hv = p * w1[c] + b1[c];
  float s1 = hv, s2 = hv * hv;
  for (int o = 16; o > 0; o >>= 1) {
    s1 += __shfl_xor(s1, o);
    s2 += __shfl_xor(s2, o);
  }
  if ((c & 31) == 0) { red[c >> 5] = s1; red[8 + (c >> 5)] = s2; }
  __syncthreads();
  float t1 = 0.f, t2 = 0.f;
  for (int i = 0; i < 8; ++i) { t1 += red[i]; t2 += red[8 + i]; }
  float mu  = t1 / (float)HID;
  float var = t2 / (float)HID - mu * mu;
  float nv  = (hv - mu) * rsqrtf(var + 1e-5f) * g1[c] + bt1[c];
  out[(size_t)row * HID + c] = (nv >= 0.f) ? nv : 0.01f * nv;
}

// ---------------------------------------------------------------------------
// Kernel 3: generic [rows x 256] @ [256 x 256] + bias, LayerNorm, leaky ReLU.
// 16-row strip per block; WMMA f32 16x16x4; in-place safe (strip staged in LDS).
// ---------------------------------------------------------------------------
__global__ __launch_bounds__(256) void mlp_gemm_kernel(
    const float* __restrict__ hin, const float* __restrict__ W,
    const float* __restrict__ bias, const float* __restrict__ gam,
    const float* __restrict__ bet, float* __restrict__ hout) {
  __shared__ float As[16][HID];
  __shared__ float Cs[16][HID];
  __shared__ float redA[16][16];
  __shared__ float redB[16][16];
  __shared__ float mus[16], vars[16];

  const int tid  = threadIdx.x;
  const int lane = tid & 31;
  const int wv   = tid >> 5;
  const int row0 = blockIdx.x * 16;

  for (int i = tid; i < 16 * HID; i += 256)
    As[i >> 8][i & 255] = hin[(size_t)(row0 + (i >> 8)) * HID + (i & 255)];
  __syncthreads();

  const int m = lane & 15;
  const int h = lane >> 4;
  for (int nt = wv * 2; nt < wv * 2 + 2; ++nt) {   // 2 N-tiles per wave
    const int n = nt * 16 + m;
    v8f acc = {};
    for (int k = 0; k < HID; k += 4) {
      __builtin_prefetch(&W[(size_t)(k + 8) * HID + n], 0, 1);
      v2f a = *(const v2f*)&As[m][k + 2 * h];
      v2f bb;
      bb.x = W[(size_t)(k + 2 * h) * HID + n];
      bb.y = W[(size_t)(k + 2 * h + 1) * HID + n];
      acc = wmma_f32(a, bb, acc);
    }
#pragma unroll
    for (int v = 0; v < 8; ++v) {
      int M = v + 8 * h;
      Cs[M][n] = acc[v] + bias[n];
    }
  }
  __syncthreads();

  // LayerNorm + leaky ReLU; 16 threads cooperate per row.
  const int r = tid >> 4, g = tid & 15;
  float s1 = 0.f, s2 = 0.f;
  for (int c = g; c < HID; c += 16) {
    float v = Cs[r][c];
    s1 += v; s2 += v * v;
  }
  redA[r][g] = s1; redB[r][g] = s2;
  __syncthreads();
  if (g == 0) {
    float t1 = 0.f, t2 = 0.f;
    for (int i = 0; i < 16; ++i) { t1 += redA[r][i]; t2 += redB[r][i]; }
    float mu = t1 / (float)HID;
    mus[r] = mu;
    vars[r] = t2 / (float)HID - mu * mu;
  }
  __syncthreads();
  float mu = mus[r], rstd = rsqrtf(vars[r] + 1e-5f);
  for (int c = g; c < HID; c += 16) {
    float v = (Cs[r][c] - mu) * rstd * gam[c] + bet[c];
    hout[(size_t)(row0 + r) * HID + c] = (v >= 0.f) ? v : 0.01f * v;
  }
}

// ---------------------------------------------------------------------------
extern "C" void kernel_launch(void* const* d_in, const int* in_sizes, int n_in,
                              void* d_out, int out_size, void* d_ws,
                              size_t ws_size, hipStream_t stream) {
  const float* y   = (const float*)d_in[0];
  const float* w1  = (const float*)d_in[1];
  const float* b1  = (const float*)d_in[2];
  const float* g1  = (const float*)d_in[3];
  const float* bt1 = (const float*)d_in[4];
  const float* w2  = (const float*)d_in[5];
  const float* b2  = (const float*)d_in[6];
  const float* g2  = (const float*)d_in[7];
  const float* bt2 = (const float*)d_in[8];
  const float* w3  = (const float*)d_in[9];
  const float* b3  = (const float*)d_in[10];
  const float* g3  = (const float*)d_in[11];
  const float* bt3 = (const float*)d_in[12];
  float* out = (float*)d_out;

  float* pitchbuf = (float*)d_ws;          // 16*391 floats
  float* hbuf     = pitchbuf + 8192;       // 6400*256 floats

  yin_kernel<<<16 * NFR_IN, 256, 0, stream>>>(y, pitchbuf);
  mlp1_kernel<<<16 * NFR_OUT, 256, 0, stream>>>(pitchbuf, w1, b1, g1, bt1, hbuf);
  mlp_gemm_kernel<<<(16 * NFR_OUT) / 16, 256, 0, stream>>>(hbuf, w2, b2, g2, bt2, hbuf);
  mlp_gemm_kernel<<<(16 * NFR_OUT) / 16, 256, 0, stream>>>(hbuf, w3, b3, g3, bt3, out);
}